// kernel_computation_90297392431225
// MI455X (gfx1250) — compile-verified
//
#include <hip/hip_runtime.h>

// Problem constants (match reference)
#define KSZ   5
#define PADR  2
#define CH    21
#define KK    25      // KSZ*KSZ
#define IMH   256
#define IMW   256
#define NB    4

// Tiling
#define TW    32
#define TH    16
#define LW    (TW + 2*PADR)   // 36
#define LH    (TH + 2*PADR)   // 20
#define NTHREADS 128          // 8 strips (4 px each) x 16 rows

// ---- CDNA5 async global->LDS path (probe-verified signatures:
//   b32: (as1 int*   gsrc, as3 int*   ldst, imm offset, imm cpol)
//   b64: (as1 v2i32* gsrc, as3 v2i32* ldst, imm offset, imm cpol) ) ----
#if defined(__has_builtin)
#  if __has_builtin(__builtin_amdgcn_global_load_async_to_lds_b64)
#    define HAVE_ASYNC_B64 1
#  endif
#  if __has_builtin(__builtin_amdgcn_global_load_async_to_lds_b32)
#    define HAVE_ASYNC_B32 1
#  endif
#  if __has_builtin(__builtin_amdgcn_s_wait_asynccnt)
#    define HAVE_WAIT_ASYNC_BUILTIN 1
#  endif
#endif

#if defined(HAVE_ASYNC_B64) || defined(HAVE_ASYNC_B32)
#  define HAVE_ASYNC_LDS 1
#endif

typedef int v2i __attribute__((ext_vector_type(2)));
typedef __attribute__((address_space(1))) int g_i32;
typedef __attribute__((address_space(3))) int s_i32;
typedef __attribute__((address_space(1))) v2i g_v2i;
typedef __attribute__((address_space(3))) v2i s_v2i;

// Stage 2 consecutive floats (8B-aligned on both sides by construction).
__device__ __forceinline__ void stage_b64(const float* g, float* l) {
#if defined(HAVE_ASYNC_B64)
    __builtin_amdgcn_global_load_async_to_lds_b64(
        (g_v2i*)const_cast<float*>(g), (s_v2i*)l, 0, 0);
#elif defined(HAVE_ASYNC_B32)
    __builtin_amdgcn_global_load_async_to_lds_b32(
        (g_i32*)const_cast<float*>(g), (s_i32*)l, 0, 0);
    __builtin_amdgcn_global_load_async_to_lds_b32(
        (g_i32*)const_cast<float*>(g + 1), (s_i32*)(l + 1), 0, 0);
#else
    l[0] = g[0];
    l[1] = g[1];
#endif
}

__device__ __forceinline__ void wait_async_done() {
#if defined(HAVE_ASYNC_LDS)
#  if defined(HAVE_WAIT_ASYNC_BUILTIN)
    __builtin_amdgcn_s_wait_asynccnt(0);
#  else
    asm volatile("s_wait_asynccnt 0" ::: "memory");
#  endif
#endif
}

__global__ __launch_bounds__(NTHREADS)
void kpn_dynconv5x5_kernel(const float* __restrict__ x,
                           const float* __restrict__ wk,
                           float* __restrict__ out) {
    __shared__ __align__(16) float lds[CH * LH * LW];   // 60,480 bytes

    const int b   = blockIdx.z;
    const int x0  = blockIdx.x * TW;
    const int y0  = blockIdx.y * TH;
    const int tid = threadIdx.x;

    // Per-thread output strip: 4 consecutive pixels
    const int s   = tid & 7;          // strip column 0..7
    const int ty  = tid >> 3;         // row 0..15
    const int gxs = x0 + s * 4;       // 16B-aligned pixel column
    const int gy  = y0 + ty;

    const float* wb = wk + (((size_t)b * KK) * IMH + gy) * IMW + gxs;

    // Prefetch this thread's 25 weight vectors (global_prefetch_b8) so the
    // weight cachelines stream in while we stage x into LDS.
#pragma unroll
    for (int k = 0; k < KK; ++k)
        __builtin_prefetch(wb + (size_t)k * IMH * IMW, 0, 0);

    // ---- Stage haloed x tile (CH x LH x LW) into LDS via async b64 copies.
    // W, x0 and the -2 halo offset are all even, so each 2-float chunk is
    // either fully in-bounds or fully out-of-bounds: no partial chunks. ----
    const float* xb = x + (size_t)b * CH * IMH * IMW;
    const int HALF = LW / 2;                       // 18 chunks per row
    for (int idx = tid; idx < CH * LH * HALF; idx += NTHREADS) {
        const int c   = idx / (LH * HALF);
        const int rem = idx - c * (LH * HALF);
        const int ry  = rem / HALF;
        const int rx  = (rem - ry * HALF) * 2;
        const int gyy = y0 - PADR + ry;
        const int gxx = x0 - PADR + rx;            // even; gxx ok => gxx+1 ok
        float* ldst = &lds[(c * LH + ry) * LW + rx];
        if ((unsigned)gyy < IMH && (unsigned)gxx < IMW) {
            stage_b64(xb + ((size_t)c * IMH + gyy) * IMW + gxx, ldst);
        } else {
            *reinterpret_cast<float2*>(ldst) = make_float2(0.f, 0.f);
        }
    }
    wait_async_done();
    __syncthreads();

    // ---- Load 25 per-pixel weight vectors (coalesced float4) ----
    float4 wv[KK];
#pragma unroll
    for (int k = 0; k < KK; ++k)
        wv[k] = *reinterpret_cast<const float4*>(wb + (size_t)k * IMH * IMW);

    float* ob = out + (((size_t)b * CH) * IMH + gy) * IMW + gxs;

    // ---- Compute: per channel, 5 rows x 2 aligned b128 LDS loads ----
    for (int c = 0; c < CH; ++c) {
        float4 acc = make_float4(0.f, 0.f, 0.f, 0.f);
        const float* lrow = &lds[((size_t)c * LH + ty) * LW + s * 4];
#pragma unroll
        for (int i = 0; i < KSZ; ++i) {
            const float4* lp = reinterpret_cast<const float4*>(lrow + i * LW);
            const float4 v0 = lp[0];
            const float4 v1 = lp[1];
            const float vbuf[8] = {v0.x, v0.y, v0.z, v0.w,
                                   v1.x, v1.y, v1.z, v1.w};
#pragma unroll
            for (int j = 0; j < KSZ; ++j) {
                const float4 wt = wv[i * KSZ + j];
                acc.x += vbuf[j + 0] * wt.x;
                acc.y += vbuf[j + 1] * wt.y;
                acc.z += vbuf[j + 2] * wt.z;
                acc.w += vbuf[j + 3] * wt.w;
            }
        }
        *reinterpret_cast<float4*>(ob + (size_t)c * IMH * IMW) = acc;
    }
}

extern "C" void kernel_launch(void* const* d_in, const int* in_sizes, int n_in,
                              void* d_out, int out_size, void* d_ws, size_t ws_size,
                              hipStream_t stream) {
    const float* x  = (const float*)d_in[0];   // coarse_prediction [4,21,256,256]
    const float* wk = (const float*)d_in[1];   // kernel_collection [4,25,256,256]
    float* out = (float*)d_out;                // [4,21,256,256]

    dim3 grid(IMW / TW, IMH / TH, NB);         // 8 x 16 x 4 = 512 blocks
    dim3 block(NTHREADS);
    hipLaunchKernelGGL(kpn_dynconv5x5_kernel, grid, block, 0, stream, x, wk, out);
}